// YoloXWrapper_807453851970
// MI455X (gfx1250) — compile-verified
//
#include <hip/hip_runtime.h>
#include <hip/hip_bf16.h>
#include <stdint.h>

// ---------------------------------------------------------------------------
// YOLOX postprocess (score -> thresh -> top512 -> NMS -> top100) + features
// passthrough, targeted at MI455X (gfx1250, wave32).
//
// Perf model: HBM-bound on the 75.5MB features copy (~3.2us @ 23.3TB/s) +
// 18.6MB pred_boxes scan. Scores (3MB) are L2-resident (192MB L2), so the
// exact 31-pass radix top-k re-reads them from L2 at negligible cost.
// CDNA5-specific paths: v_wmma_f32_16x16x4_f32 for the pairwise area-sum
// term of the IOU denominator, and batched async global<->LDS b128 DMA for
// the bulk copy (4 in-flight b128 per lane to cover HBM latency).
// IOU threshold test is divide-free: den > 0, so iou>T <=> inter > T*den.
// ---------------------------------------------------------------------------

#define B_ 16
#define N_ 48384
#define K_ 512
#define MAXDET_ 100
#define IOU_T 0.45f

typedef float v2f __attribute__((ext_vector_type(2)));
typedef float v8f __attribute__((ext_vector_type(8)));

// ---------------------------------------------------------------- init ws ---
__global__ void init_kernel(unsigned* wmax) {
  if (threadIdx.x < B_) wmax[threadIdx.x] = 0u;
}

// ------------------------------------------------------- scores + per-b max -
// 189 blocks per image (N divisible by 256), block-reduced max -> atomicMax.
__global__ __launch_bounds__(256) void score_kernel(const float* __restrict__ pred,
                                                    float* __restrict__ scores,
                                                    unsigned* __restrict__ wmax) {
  __shared__ unsigned red[256];
  int gid = blockIdx.x * 256 + threadIdx.x;       // < B*N = 774144
  const float* p = pred + (size_t)gid * 6;
  float s = p[4] * p[5];                          // obj * class_conf (1 class)
  scores[gid] = s;
  red[threadIdx.x] = __float_as_uint(s);          // s >= 0 -> uint-monotonic
  __syncthreads();
  for (int st = 128; st > 0; st >>= 1) {
    if (threadIdx.x < st) {
      unsigned a = red[threadIdx.x], b = red[threadIdx.x + st];
      red[threadIdx.x] = a > b ? a : b;
    }
    __syncthreads();
  }
  if (threadIdx.x == 0) {
    int b = blockIdx.x / (N_ / 256);              // blocks never straddle images
    atomicMax(&wmax[b], red[0]);
  }
}

// ----------------------------------------------------------------- thresh ---
// t=0.25; while count(s>=t)==0: t=max(t-0.1, t/10)  <=>  while max<t.
__global__ void thresh_kernel(const unsigned* __restrict__ wmax,
                              float* __restrict__ thresh) {
  int b = threadIdx.x;
  if (b < B_) {
    float m = __uint_as_float(wmax[b]);
    float t = 0.25f;
    for (int it = 0; it < 64 && m < t; ++it) t = fmaxf(t - 0.1f, t / 10.0f);
    thresh[b] = t;
  }
}

// ----------------------------------------------------- exact top-512 per b --
// Radix-select the 512th-largest masked score bit pattern, then compact
// (strict-greater via scan, ties filled lowest-index-first via contiguous
// per-thread chunks + block scan), then bitonic-sort 512 keys
// (key = (~scorebits)<<32 | idx  => ascending key == score desc, idx asc,
// matching lax.top_k tie-breaking).
__device__ inline void scan1024(unsigned* sc, int tid, unsigned val,
                                unsigned& excl, unsigned& total) {
  sc[tid] = val;
  __syncthreads();
  for (int off = 1; off < 1024; off <<= 1) {
    unsigned add = (tid >= off) ? sc[tid - off] : 0u;
    __syncthreads();
    sc[tid] += add;
    __syncthreads();
  }
  excl = sc[tid] - val;
  total = sc[1023];
  __syncthreads();
}

__global__ __launch_bounds__(1024) void topk_kernel(const float* __restrict__ scores,
                                                    const float* __restrict__ thresh,
                                                    float* __restrict__ cscore,
                                                    int* __restrict__ cidx) {
  __shared__ unsigned red[1024];
  __shared__ unsigned sc[1024];
  __shared__ unsigned long long keys[K_];
  int b = blockIdx.x, tid = threadIdx.x;
  const float thr = thresh[b];
  const float* S = scores + (size_t)b * N_;
  const int CH = (N_ + 1023) / 1024;              // 48, contiguous chunk/thread
  int n0 = tid * CH;
  int n1 = n0 + CH; if (n1 > N_) n1 = N_;

  // radix select: p = 512th largest masked bit pattern
  unsigned p = 0;
  for (int bit = 30; bit >= 0; --bit) {
    unsigned q = p | (1u << bit);
    unsigned c = 0;
    for (int n = n0; n < n1; ++n) {
      float s = S[n];
      unsigned u = (s >= thr) ? __float_as_uint(s) : 0u;
      c += (u >= q) ? 1u : 0u;
    }
    red[tid] = c;
    __syncthreads();
    for (int st = 512; st > 0; st >>= 1) {
      if (tid < st) red[tid] += red[tid + st];
      __syncthreads();
    }
    if (red[0] >= K_) p = q;                      // uniform decision
    __syncthreads();
  }

  // per-thread strict-greater / tie counts over contiguous index ranges
  unsigned gc = 0, tc = 0;
  for (int n = n0; n < n1; ++n) {
    float s = S[n];
    unsigned u = (s >= thr) ? __float_as_uint(s) : 0u;
    gc += (u > p) ? 1u : 0u;
    tc += (u == p) ? 1u : 0u;
  }
  unsigned gExcl, TG, tExcl, tTot;
  scan1024(sc, tid, gc, gExcl, TG);               // TG = count(u > p) < 512
  scan1024(sc, tid, tc, tExcl, tTot);

  // fill candidate list
  unsigned g = gExcl, t = tExcl;
  for (int n = n0; n < n1; ++n) {
    float s = S[n];
    unsigned u = (s >= thr) ? __float_as_uint(s) : 0u;
    if (u > p) {
      keys[g++] = (((unsigned long long)(~u)) << 32) | (unsigned)n;
    } else if (u == p) {
      if (TG + t < K_)                            // lowest-index ties win
        keys[TG + t] = (((unsigned long long)(~u)) << 32) | (unsigned)n;
      ++t;
    }
  }
  __syncthreads();

  // bitonic sort 512 keys ascending
  for (int k = 2; k <= K_; k <<= 1) {
    for (int j = k >> 1; j > 0; j >>= 1) {
      if (tid < K_) {
        int ixj = tid ^ j;
        if (ixj > tid) {
          bool up = ((tid & k) == 0);
          unsigned long long a = keys[tid], c2 = keys[ixj];
          bool sw = up ? (a > c2) : (a < c2);
          if (sw) { keys[tid] = c2; keys[ixj] = a; }
        }
      }
      __syncthreads();
    }
  }

  if (tid < K_) {
    unsigned long long kk = keys[tid];
    unsigned u = ~(unsigned)(kk >> 32);
    cscore[b * K_ + tid] = __uint_as_float(u);    // masked score (0 if invalid)
    cidx[b * K_ + tid] = (int)(unsigned)kk;
  }
}

// ------------------------------------------------------------ NMS + output --
// One block (8 waves) per image. Suppression matrix (512x512 bits, 32KB LDS)
// built tile-wise: v_wmma_f32_16x16x4_f32 computes the pairwise denominator
// base area_i + area_j as the outer product [area_i,1]x[1;area_j]; lanes then
// fuse intersection, divide-free IOU compare and bit-pack. Greedy pass is
// bitmask-serial.
__global__ __launch_bounds__(256) void nms_kernel(const float* __restrict__ pred,
                                                  const float* __restrict__ cscore,
                                                  const int* __restrict__ cidx,
                                                  float* __restrict__ det,
                                                  float* __restrict__ maskout) {
  __shared__ float lx1[K_], ly1[K_], lx2[K_], ly2[K_];
  __shared__ float larea[K_], lobj[K_], lconf[K_], lsc[K_];
  __shared__ unsigned sup[K_ * 16];               // 512 rows x 512 bits
  __shared__ unsigned keepw[16];

  int b = blockIdx.x, tid = threadIdx.x;

  // gather candidates, convert cxcywh -> xyxy, precompute areas
  for (int c = tid; c < K_; c += 256) {
    int idx = cidx[b * K_ + c];
    const float* p = pred + ((size_t)b * N_ + (size_t)idx) * 6;
    float cx = p[0], cy = p[1], w = p[2], h = p[3];
    float x1 = cx - 0.5f * w, y1 = cy - 0.5f * h;
    float x2 = cx + 0.5f * w, y2 = cy + 0.5f * h;
    lx1[c] = x1; ly1[c] = y1; lx2[c] = x2; ly2[c] = y2;
    larea[c] = fmaxf(x2 - x1, 0.0f) * fmaxf(y2 - y1, 0.0f);
    lobj[c] = p[4]; lconf[c] = p[5];
    lsc[c] = cscore[b * K_ + c];
  }
  for (int i = tid; i < K_ * 16; i += 256) sup[i] = 0u;
  if (tid < 16) keepw[tid] = 0u;
  __syncthreads();
  for (int c = tid; c < K_; c += 256)
    if (lsc[c] > 0.0f) atomicOr(&keepw[c >> 5], 1u << (c & 31));

  // suppression matrix: 32x32 grid of 16x16 tiles, 8 waves round-robin
  int wave = tid >> 5, lane = tid & 31, half = lane >> 4, l = lane & 15;
  for (int tile = wave; tile < 32 * 32; tile += 8) {
    int i0 = (tile >> 5) * 16, j0 = (tile & 31) * 16;
    // A (16x4): row m = [area_i, 1, 0, 0]; lanes0-15 hold K=0,1; 16-31 K=2,3
    v2f a; a.x = half ? 0.0f : larea[i0 + l]; a.y = half ? 0.0f : 1.0f;
    // B (4x16): K0 row = ones, K1 row = area_j, K2/K3 = 0
    v2f bm; bm.x = half ? 0.0f : 1.0f; bm.y = half ? 0.0f : larea[j0 + l];
    v8f c0 = {0.f, 0.f, 0.f, 0.f, 0.f, 0.f, 0.f, 0.f};
    v8f d = __builtin_amdgcn_wmma_f32_16x16x4_f32(
        false, a, false, bm, (short)0, c0, false, false);
    // D[m][n]: lane l (+16*half), VGPR k -> (i = i0+k+8*half, j = j0+l)
    int j = j0 + l;
    float jx1 = lx1[j], jy1 = ly1[j], jx2 = lx2[j], jy2 = ly2[j];
    for (int k = 0; k < 8; ++k) {
      int i = i0 + k + (half ? 8 : 0);
      float xx1 = fmaxf(lx1[i], jx1), yy1 = fmaxf(ly1[i], jy1);
      float xx2 = fminf(lx2[i], jx2), yy2 = fminf(ly2[i], jy2);
      float inter = fmaxf(xx2 - xx1, 0.0f) * fmaxf(yy2 - yy1, 0.0f);
      // iou = inter/(d[k]-inter+1e-9) > T  <=>  inter > T*(d[k]-inter+1e-9)
      // (denominator strictly positive: inter <= min(area_i, area_j))
      float den = d[k] - inter + 1e-9f;
      if (inter > IOU_T * den) atomicOr(&sup[i * 16 + (j >> 5)], 1u << (j & 31));
    }
  }
  __syncthreads();

  // greedy NMS: serial over i, vector-of-bits updates
  if (tid == 0) {
    for (int i = 0; i < K_; ++i) {
      if ((keepw[i >> 5] >> (i & 31)) & 1u) {
        int wi = i >> 5, bp = i & 31;
        unsigned mgt = (bp < 31) ? (0xFFFFFFFFu << (bp + 1)) : 0u;
        keepw[wi] &= ~(sup[i * 16 + wi] & mgt);
        for (int w = wi + 1; w < 16; ++w) keepw[w] &= ~sup[i * 16 + w];
      }
    }
  }
  __syncthreads();

  // emit det/mask: candidates already score-sorted, so top-100 of kept ==
  // first <=100 kept candidates in order; remaining rows zero.
  float* detb = det + (size_t)b * MAXDET_ * 7;
  float* mb = maskout + (size_t)b * MAXDET_;
  for (int r = tid; r < MAXDET_; r += 256) {
    for (int q = 0; q < 7; ++q) detb[r * 7 + q] = 0.0f;
    mb[r] = 0.0f;
  }
  __syncthreads();
  if (tid == 0) {
    int cnt = 0;
    for (int c = 0; c < K_ && cnt < MAXDET_; ++c) {
      if ((keepw[c >> 5] >> (c & 31)) & 1u) {
        float* row = detb + cnt * 7;
        row[0] = lx1[c]; row[1] = ly1[c]; row[2] = lx2[c]; row[3] = ly2[c];
        row[4] = lobj[c]; row[5] = lconf[c]; row[6] = 0.0f;  // class id = 0
        mb[cnt] = 1.0f;
        ++cnt;
      }
    }
  }
}

// ------------------------------------------------ features copy (async DMA) -
// Each thread owns exactly 4 b128 vectors (wave-contiguous 512B per issue).
// All 4 async loads are issued back-to-back (4 in-flight DMAs per lane to
// cover HBM latency), one asynccnt wait, then 4 async stores, one wait.
// grid = nvec16 / 1024 blocks of 256 threads (exact for this shape).
__global__ __launch_bounds__(256) void copy_features_kernel(const float* __restrict__ src,
                                                            float* __restrict__ dst) {
  __shared__ char buf[256 * 4 * 16];
  int tid = threadIdx.x;
  unsigned long long v0 = (unsigned long long)blockIdx.x * 1024ull + (unsigned)tid;
  unsigned long long sbase = (unsigned long long)(uintptr_t)src;
  unsigned long long dbase = (unsigned long long)(uintptr_t)dst;
#pragma unroll
  for (int k = 0; k < 4; ++k) {
    __attribute__((address_space(3))) char* lp =
        (__attribute__((address_space(3))) char*)&buf[(k * 256 + tid) * 16];
    unsigned long long sa = sbase + (v0 + (unsigned long long)(k * 256)) * 16ull;
    asm volatile("global_load_async_to_lds_b128 %0, %1, off"
                 :: "v"(lp), "v"(sa) : "memory");
  }
  asm volatile("s_wait_asynccnt 0" ::: "memory");
#pragma unroll
  for (int k = 0; k < 4; ++k) {
    __attribute__((address_space(3))) char* lp =
        (__attribute__((address_space(3))) char*)&buf[(k * 256 + tid) * 16];
    unsigned long long da = dbase + (v0 + (unsigned long long)(k * 256)) * 16ull;
    asm volatile("global_store_async_from_lds_b128 %0, %1, off"
                 :: "v"(da), "v"(lp) : "memory");
  }
  asm volatile("s_wait_asynccnt 0" ::: "memory");
}

// scalar tail for sizes not divisible by 4096 floats (none for this shape)
__global__ void copy_tail_kernel(const float* __restrict__ src,
                                 float* __restrict__ dst, int start, int total) {
  int i = start + blockIdx.x * 256 + threadIdx.x;
  if (i < total) dst[i] = src[i];
}

// ---------------------------------------------------------------------------
extern "C" void kernel_launch(void* const* d_in, const int* in_sizes, int n_in,
                              void* d_out, int out_size, void* d_ws, size_t ws_size,
                              hipStream_t stream) {
  const float* pred = (const float*)d_in[0];      // (16, 48384, 6)
  const float* feat = (const float*)d_in[1];      // (16, 256, 48, 48)

  // workspace layout
  char* ws = (char*)d_ws;
  float* ws_scores = (float*)ws;                               // B*N f32
  size_t off = (size_t)B_ * N_ * sizeof(float);
  unsigned* ws_max = (unsigned*)(ws + off);  off += 256;
  float* ws_thresh = (float*)(ws + off);     off += 256;
  float* ws_cscore = (float*)(ws + off);     off += (size_t)B_ * K_ * sizeof(float);
  int* ws_cidx = (int*)(ws + off);

  // output layout: det (16*100*7) | mask (16*100) | features (16*256*48*48)
  float* out = (float*)d_out;
  float* det = out;
  float* maskout = out + B_ * MAXDET_ * 7;
  float* feat_out = out + B_ * MAXDET_ * 7 + B_ * MAXDET_;

  init_kernel<<<1, 32, 0, stream>>>(ws_max);
  score_kernel<<<(B_ * N_) / 256, 256, 0, stream>>>(pred, ws_scores, ws_max);
  thresh_kernel<<<1, 32, 0, stream>>>(ws_max, ws_thresh);
  topk_kernel<<<B_, 1024, 0, stream>>>(ws_scores, ws_thresh, ws_cscore, ws_cidx);
  nms_kernel<<<B_, 256, 0, stream>>>(pred, ws_cscore, ws_cidx, det, maskout);

  int nfeat = in_sizes[1];                        // 9,437,184 floats
  int nvec16 = nfeat / 4;                         // 2,359,296 x 16B
  int nblocks = nvec16 / 1024;                    // 2304 (exact here)
  if (nblocks > 0)
    copy_features_kernel<<<nblocks, 256, 0, stream>>>(feat, feat_out);
  int done = nblocks * 4096;                      // floats covered by DMA path
  int rem = nfeat - done;
  if (rem > 0)
    copy_tail_kernel<<<(rem + 255) / 256, 256, 0, stream>>>(feat, feat_out, done, nfeat);
}